// OHEMBCEWithLogitsLoss_11347303596649
// MI455X (gfx1250) — compile-verified
//
#include <hip/hip_runtime.h>
#include <hip/hip_bf16.h>

typedef __attribute__((ext_vector_type(2))) float v2f;
typedef __attribute__((ext_vector_type(8))) float v8f;

#define NWG   512
#define TPB   256
#define NB1   4096          // level-1 bins over bits[30:19]
#define SH1   19
#define NB2   4096          // refine bins over bits[18:7]
#define SH2   7
// ws layout in 32-bit words
#define IDX_HCNT 0          // 4096 u32
#define IDX_HSUM 4096       // 4096 f32
#define IDX_SCNT 8192       // 4096 u32
#define IDX_SSUM 12288      // 4096 f32
#define IDX_NPOS 16384      // u32
#define IDX_K    16385      // u32
#define IDX_T    16386      // u32
#define IDX_CAB  16387      // u32 (count strictly above T)
#define IDX_SAB  16388      // f32 (sum strictly above T)
#define IDX_NEED 16389      // u32 (k - C_above)
#define IDX_PART 16392      // NWG f32 pos-sum partials
#define ZWORDS   (IDX_PART + NWG)
#define NOBIN    0xFFFFFFFFu

__device__ __forceinline__ float softplus_tail(float ax) {
    // log1p(exp(-|x|)); exp arg <= 0 so no overflow
    return log1pf(__expf(-ax));
}

__global__ void k_zero(unsigned* __restrict__ ws, int nwords) {
    int i = blockIdx.x * blockDim.x + threadIdx.x;
    if (i < nwords) ws[i] = 0u;
}

// Pass 1: stream all elements; positives -> pos count/sum; negatives -> 4096-bin
// histogram (counts + float sums) over high bits of bce's IEEE pattern.
__global__ void __launch_bounds__(TPB)
k_pass1(const float* __restrict__ logits, const float* __restrict__ target,
        const float* __restrict__ pw_p, unsigned* __restrict__ wsu,
        float* __restrict__ wsf, unsigned n, unsigned sp_shift) {
    __shared__ unsigned s_cnt[NB1];
    __shared__ float    s_sum[NB1];
    __shared__ float    s_red[TPB];
    __shared__ unsigned s_redc[TPB];

    for (int i = threadIdx.x; i < NB1; i += blockDim.x) { s_cnt[i] = 0u; s_sum[i] = 0.0f; }
    __syncthreads();

    const float pw = pw_p[0];
    const unsigned sp_mask = (1u << sp_shift) - 1u;
    const unsigned stride  = gridDim.x * blockDim.x * 4u;
    unsigned e = (blockIdx.x * blockDim.x + threadIdx.x) * 4u;

    float    pos_acc = 0.0f;
    unsigned pos_cnt = 0u;

    for (; e < n; e += stride) {
        unsigned b  = e >> sp_shift;
        unsigned li = e + ((b + 1u) << sp_shift);   // == (2b+1)*SP + sp
        // prefetch next strip (gfx1250 global_prefetch_b8; speculative, safe OOB)
        {
            unsigned en = e + stride;
            unsigned bn = en >> sp_shift;
            __builtin_prefetch(target + en, 0, 1);
            __builtin_prefetch(logits + en + ((bn + 1u) << sp_shift), 0, 1);
        }
        float4 tv = *reinterpret_cast<const float4*>(target + e);
        float4 xv = *reinterpret_cast<const float4*>(logits + li);
        float xs[4] = {xv.x, xv.y, xv.z, xv.w};
        float ts[4] = {tv.x, tv.y, tv.z, tv.w};
#pragma unroll
        for (int j = 0; j < 4; ++j) {
            float x  = xs[j];
            float su = softplus_tail(fabsf(x));
            if (ts[j] > 0.5f) {
                pos_acc += pw * (fmaxf(-x, 0.0f) + su);
                pos_cnt++;
            } else {
                float bce = fmaxf(x, 0.0f) + su;           // >= 0 -> bits monotonic
                unsigned bin = __float_as_uint(bce) >> SH1; // 0..4095
                atomicAdd(&s_cnt[bin], 1u);
                atomicAdd(&s_sum[bin], bce);
            }
        }
    }

    s_red[threadIdx.x]  = pos_acc;
    s_redc[threadIdx.x] = pos_cnt;
    __syncthreads();

    // flush private histogram to global
    for (int i = threadIdx.x; i < NB1; i += blockDim.x) {
        unsigned c = s_cnt[i];
        if (c) {
            atomicAdd(&wsu[IDX_HCNT + i], c);
            atomicAdd(&wsf[IDX_HSUM + i], s_sum[i]);
        }
    }
    // workgroup reduction of pos stats
    for (int off = TPB / 2; off > 0; off >>= 1) {
        if ((int)threadIdx.x < off) {
            s_red[threadIdx.x]  += s_red[threadIdx.x + off];
            s_redc[threadIdx.x] += s_redc[threadIdx.x + off];
        }
        __syncthreads();
    }
    if (threadIdx.x == 0) {
        atomicAdd(&wsu[IDX_NPOS], s_redc[0]);          // integer: deterministic
        wsf[IDX_PART + blockIdx.x] = s_red[0];         // per-WG slot, reduced later in order
    }
}

// Scan: find k, threshold bin T, exact count/sum strictly above T.
__global__ void k_scan(unsigned* __restrict__ wsu, float* __restrict__ wsf, unsigned n) {
    __shared__ unsigned c_cnt[256];
    __shared__ float    c_sum[256];
    int tid = threadIdx.x;
    unsigned mc = 0u; float ms = 0.0f;
    for (int i = 0; i < 16; ++i) {
        int b = tid * 16 + i;
        mc += wsu[IDX_HCNT + b];
        ms += wsf[IDX_HSUM + b];
    }
    c_cnt[tid] = mc; c_sum[tid] = ms;
    __syncthreads();
    if (tid == 0) {
        unsigned num_pos = wsu[IDX_NPOS];
        unsigned num_neg = n - num_pos;
        unsigned k;
        if (num_pos > 0u) {
            unsigned kk = 20u * num_pos;
            k = (kk < num_neg) ? kk : num_neg;
        } else {
            unsigned kk = num_neg / 100u;
            k = (kk > 1u) ? kk : 1u;
        }
        unsigned T = NOBIN, C_above = 0u;
        float S_above = 0.0f;
        if (num_neg != 0u && k != 0u) {
            unsigned cum = 0u; float ssum = 0.0f; int found = 0;
            for (int c = 255; c >= 0 && !found; --c) {
                if (cum + c_cnt[c] >= k) {
                    for (int b = c * 16 + 15; b >= c * 16; --b) {
                        unsigned cb = wsu[IDX_HCNT + b];
                        if (cum + cb >= k) { T = (unsigned)b; found = 1; break; }
                        cum += cb; ssum += wsf[IDX_HSUM + b];
                    }
                } else {
                    cum += c_cnt[c]; ssum += c_sum[c];
                }
            }
            C_above = cum; S_above = ssum;
        } else {
            k = 0u;
        }
        wsu[IDX_K]    = k;
        wsu[IDX_T]    = T;
        wsu[IDX_CAB]  = C_above;
        wsf[IDX_SAB]  = S_above;
        wsu[IDX_NEED] = k - C_above;
    }
}

// Pass 2: sub-histogram (next 12 mantissa bits) of negatives landing in bin T.
__global__ void __launch_bounds__(TPB)
k_pass2(const float* __restrict__ logits, const float* __restrict__ target,
        unsigned* __restrict__ wsu, float* __restrict__ wsf,
        unsigned n, unsigned sp_shift) {
    const unsigned T = wsu[IDX_T];
    if (T > (NB1 - 1u)) return;   // no refinement needed (uniform scalar branch)

    __shared__ unsigned s_cnt[NB2];
    __shared__ float    s_sum[NB2];
    for (int i = threadIdx.x; i < NB2; i += blockDim.x) { s_cnt[i] = 0u; s_sum[i] = 0.0f; }
    __syncthreads();

    const unsigned sp_mask = (1u << sp_shift) - 1u;
    const unsigned stride  = gridDim.x * blockDim.x * 4u;
    unsigned e = (blockIdx.x * blockDim.x + threadIdx.x) * 4u;

    for (; e < n; e += stride) {
        unsigned b  = e >> sp_shift;
        unsigned li = e + ((b + 1u) << sp_shift);
        {
            unsigned en = e + stride;
            unsigned bn = en >> sp_shift;
            __builtin_prefetch(target + en, 0, 1);
            __builtin_prefetch(logits + en + ((bn + 1u) << sp_shift), 0, 1);
        }
        float4 tv = *reinterpret_cast<const float4*>(target + e);
        float4 xv = *reinterpret_cast<const float4*>(logits + li);
        float xs[4] = {xv.x, xv.y, xv.z, xv.w};
        float ts[4] = {tv.x, tv.y, tv.z, tv.w};
#pragma unroll
        for (int j = 0; j < 4; ++j) {
            if (ts[j] <= 0.5f) {
                float x   = xs[j];
                float bce = fmaxf(x, 0.0f) + softplus_tail(fabsf(x));
                unsigned bits = __float_as_uint(bce);
                if ((bits >> SH1) == T) {
                    unsigned sub = (bits >> SH2) & (NB2 - 1u);
                    atomicAdd(&s_cnt[sub], 1u);
                    atomicAdd(&s_sum[sub], bce);
                }
            }
        }
    }
    __syncthreads();
    for (int i = threadIdx.x; i < NB2; i += blockDim.x) {
        unsigned c = s_cnt[i];
        if (c) {
            atomicAdd(&wsu[IDX_SCNT + i], c);
            atomicAdd(&wsf[IDX_SSUM + i], s_sum[i]);
        }
    }
}

// Finalize (one wave32): WMMA row-sum reduction of the NWG pos-sum partials
// (A = partials 16x4, B = ones => D rows = partial sums), then walk sub-hist.
__global__ void __launch_bounds__(32)
k_final(const unsigned* __restrict__ wsu, const float* __restrict__ wsf,
        float* __restrict__ out, unsigned n) {
    __shared__ unsigned fc[NB2];
    __shared__ float    fs[NB2];
    const int lane = threadIdx.x;

    // ---- WMMA reduction of NWG partials (all 32 lanes, EXEC full) ----
    v8f acc = {0.f, 0.f, 0.f, 0.f, 0.f, 0.f, 0.f, 0.f};
    v2f ones; ones.x = 1.0f; ones.y = 1.0f;
#pragma unroll
    for (int it = 0; it < NWG / 64; ++it) {
        v2f a;
        a.x = wsf[IDX_PART + it * 64 + lane * 2 + 0];
        a.y = wsf[IDX_PART + it * 64 + lane * 2 + 1];
        acc = __builtin_amdgcn_wmma_f32_16x16x4_f32(
            false, a, false, ones, (short)0, acc, false, false);
    }
    float s = acc[0] + acc[1] + acc[2] + acc[3] + acc[4] + acc[5] + acc[6] + acc[7];
    // column 0 of D: lanes 0..15 hold rowsums M=0..7 summed, lanes 16..31 M=8..15
    float pos_total = __uint_as_float(__builtin_amdgcn_readlane(__float_as_uint(s), 0)) +
                      __uint_as_float(__builtin_amdgcn_readlane(__float_as_uint(s), 16));

    // cooperative load of sub-histogram into LDS
    for (int i = lane; i < NB2; i += 32) { fc[i] = wsu[IDX_SCNT + i]; fs[i] = wsf[IDX_SSUM + i]; }
    __syncthreads();

    if (lane == 0) {
        unsigned num_pos = wsu[IDX_NPOS];
        unsigned num_neg = n - num_pos;
        unsigned k       = wsu[IDX_K];
        unsigned T       = wsu[IDX_T];
        unsigned need    = wsu[IDX_NEED];
        float    S_above = wsf[IDX_SAB];

        float pos_keep = (num_pos > 0u) ? pos_total / (float)num_pos : 0.0f;
        float neg_keep = 0.0f;
        if (num_neg > 0u && k > 0u) {
            float refined = 0.0f;
            unsigned r = need;
            if (T <= (NB1 - 1u)) {
                for (int b = NB2 - 1; b >= 0 && r > 0u; --b) {
                    unsigned c = fc[b];
                    if (!c) continue;
                    float ss = fs[b];
                    if (c <= r) { refined += ss; r -= c; }
                    else        { refined += ss * ((float)r / (float)c); r = 0u; }
                }
            }
            neg_keep = (S_above + refined) / (float)k;
        }
        out[0] = pos_keep + neg_keep;
    }
}

extern "C" void kernel_launch(void* const* d_in, const int* in_sizes, int n_in,
                              void* d_out, int out_size, void* d_ws, size_t ws_size,
                              hipStream_t stream) {
    const float* logits = (const float*)d_in[0];   // (32, 2, H, W)
    const float* target = (const float*)d_in[1];   // (32, H, W)
    const float* pw     = (const float*)d_in[2];   // scalar
    const unsigned n = (unsigned)in_sizes[1];      // 32*H*W (power of two, mult of 4)
    unsigned sp = n / 32u;                         // H*W per batch (power of two)
    unsigned sp_shift = 0u;
    while ((1u << sp_shift) < sp) sp_shift++;

    unsigned* wsu = (unsigned*)d_ws;
    float*    wsf = (float*)d_ws;

    k_zero <<<(ZWORDS + 255) / 256, 256, 0, stream>>>(wsu, ZWORDS);
    k_pass1<<<NWG, TPB, 0, stream>>>(logits, target, pw, wsu, wsf, n, sp_shift);
    k_scan <<<1, 256, 0, stream>>>(wsu, wsf, n);
    k_pass2<<<NWG, TPB, 0, stream>>>(logits, target, wsu, wsf, n, sp_shift);
    k_final<<<1, 32, 0, stream>>>(wsu, wsf, (float*)d_out, n);
}